// emdModule_53326313947396
// MI455X (gfx1250) — compile-verified
//
#include <hip/hip_runtime.h>

// Auction EMD for MI455X (gfx1250, wave32).
// value(i,j) = -(|x_i - y_j|^2 + price_j); |x_i|^2 is constant over j so it
// cancels from argmax_j and from bid = top1-top2+eps. Remaining
// f(i,j) = 2*x_i.y_j - (|y_j|^2 + price_j) is a rank-4 product:
//   A row i = (2x0, 2x1, 2x2, 1)   B col j = (y0, y1, y2, -(|y_j|^2+price_j))
// => one V_WMMA_F32_16X16X4_F32 produces an exact f32 16x16 tile of f(i,j).
// Top-2 tracking is pure compare+select (v_cmp/v_cndmask) — no branches and
// no v_max_num NaN-canonicalization chains.

typedef __attribute__((ext_vector_type(2))) float v2f;
typedef __attribute__((ext_vector_type(8))) float v8f;

#define BATCH 8
#define NPTS  2048
#define BN    (BATCH * NPTS)
#define NEGF  (-1e30f)
#define ITERS 20

// ---------------------------------------------------------------- init ----
__global__ __launch_bounds__(256) void init_kernel(
    const float* __restrict__ xyz2, const float* __restrict__ price_in,
    float4* __restrict__ ypack, float* __restrict__ price,
    float* __restrict__ max_inc, int* __restrict__ assign,
    int* __restrict__ ainv, int* __restrict__ winner)
{
  int idx = blockIdx.x * blockDim.x + threadIdx.x;
  if (idx >= BN) return;
  float y0 = xyz2[idx * 3 + 0];
  float y1 = xyz2[idx * 3 + 1];
  float y2 = xyz2[idx * 3 + 2];
  ypack[idx]   = make_float4(y0, y1, y2, y0 * y0 + y1 * y1 + y2 * y2);
  price[idx]   = price_in[idx];
  max_inc[idx] = NEGF;
  assign[idx]  = -1;
  ainv[idx]    = -1;
  winner[idx]  = NPTS;
}

// ------------------------------------------------------------ bid phase ----
__device__ __forceinline__ void top2_merge(float& v1, int& j1, float& v2,
                                           float ov1, int oj1, float ov2)
{
  // both inputs satisfy v2<=v1; tie-break argmax toward smaller j; select-only
  bool  ow   = (ov1 > v1) || ((ov1 == v1) && (oj1 < j1));
  float lose = ow ? v1 : ov1;   // loser's best
  float keep = ow ? ov2 : v2;   // winner's second
  v1 = ow ? ov1 : v1;
  j1 = ow ? oj1 : j1;
  v2 = (keep > lose) ? keep : lose;
}

__global__ __launch_bounds__(256) void bid_kernel(
    const float*  __restrict__ xyz1, const float4* __restrict__ ypack,
    const float*  __restrict__ price, const int*   __restrict__ assign,
    float* __restrict__ bidv, int* __restrict__ bestj,
    float* __restrict__ max_inc, const float* __restrict__ eps_p)
{
  const int wave = blockIdx.x * (blockDim.x >> 5) + (threadIdx.x >> 5);
  const int lane = threadIdx.x & 31;
  const int half = lane >> 4;   // K half for A/B operands; row half for C
  const int l16  = lane & 15;
  const int b       = wave / (NPTS / 16);
  const int rowBase = (wave % (NPTS / 16)) * 16;
  const int myRow   = rowBase + l16;

  // Skip whole wave (uniformly -> EXEC stays full for WMMA) if all 16 rows
  // are already assigned. Both lane halves test the same 16 rows.
  bool rowAssigned = assign[b * NPTS + myRow] >= 0;
  unsigned long long amask = __ballot(rowAssigned);
  if ((amask & 0xffffffffull) == 0xffffffffull) return;

  const float eps = eps_p[0];

  // A operand (16x4 f32): lane = M, VGPR0/1 = K0,K1 (lanes 0-15) / K2,K3 (16-31)
  float x0 = xyz1[(b * NPTS + myRow) * 3 + 0];
  float x1 = xyz1[(b * NPTS + myRow) * 3 + 1];
  float x2 = xyz1[(b * NPTS + myRow) * 3 + 2];
  v2f a;
  a.x = half ? (2.0f * x2) : (2.0f * x0);
  a.y = half ? 1.0f        : (2.0f * x1);

  // per-lane running top-2 for 8 rows (this lane's 128-column stripe)
  float v1[8], v2_[8];
  int   j1[8];
#pragma unroll
  for (int r = 0; r < 8; ++r) { v1[r] = -3.0e38f; v2_[r] = -3.0e38f; j1[r] = NPTS; }

  const float4* ypB = ypack + b * NPTS;
  const float*  prB = price + b * NPTS;

  // software pipeline: prefetch chunk c+1 while consuming chunk c
  float4 yp = ypB[l16];
  float  pj = prB[l16];
  for (int c = 0; c < NPTS / 16; ++c) {
    const int j  = c * 16 + l16;
    const int jn = j + 16;
    float4 ypn;
    float  pjn;
    if (c + 1 < NPTS / 16) { ypn = ypB[jn]; pjn = prB[jn]; }
    else                   { ypn = yp;      pjn = pj;      }

    v2f bb;  // B operand (4x16 f32): lane = N, VGPR0/1 = K0,K1 / K2,K3 per half
    bb.x = half ? yp.z           : yp.x;
    bb.y = half ? (-(yp.w + pj)) : yp.y;
    v8f acc = {};
    acc = __builtin_amdgcn_wmma_f32_16x16x4_f32(
        /*neg_a=*/false, a, /*neg_b=*/false, bb,
        /*c_mod=*/(short)0, acc, /*reuse_a=*/false, /*reuse_b=*/false);

#pragma unroll
    for (int r = 0; r < 8; ++r) {       // C: VGPR r -> row rowBase + half*8 + r
      float v   = acc[r];
      bool  gt1 = v > v1[r];            // 2 cmp + 4 cndmask, no max/canonicalize
      bool  gt2 = v > v2_[r];
      v2_[r] = gt1 ? v1[r] : (gt2 ? v : v2_[r]);
      v1[r]  = gt1 ? v : v1[r];
      j1[r]  = gt1 ? j : j1[r];
    }
    yp = ypn;
    pj = pjn;
  }

  // cross-lane top-2 tournament within each 16-lane half, then write + scatter-max
#pragma unroll
  for (int r = 0; r < 8; ++r) {
    float mv1 = v1[r]; int mj1 = j1[r]; float mv2 = v2_[r];
#pragma unroll
    for (int m = 1; m < 16; m <<= 1) {
      float ov1 = __shfl_xor(mv1, m, 32);
      int   oj1 = __shfl_xor(mj1, m, 32);
      float ov2 = __shfl_xor(mv2, m, 32);
      top2_merge(mv1, mj1, mv2, ov1, oj1, ov2);
    }
    if (l16 == r) {
      const int row = rowBase + half * 8 + r;
      const int gi  = b * NPTS + row;
      const bool assigned = assign[gi] >= 0;
      float bd = assigned ? NEGF : (mv1 - mv2 + eps);  // bd >= eps > 0 if bidding
      bidv[gi]  = bd;
      bestj[gi] = mj1;
      if (!assigned) {
        // positive floats are order-preserving under signed-int compare
        atomicMax((int*)&max_inc[b * NPTS + mj1], __float_as_int(bd));
      }
    }
  }
}

// --------------------------------------------------------- winner phase ----
__global__ __launch_bounds__(256) void winner_kernel(
    const int* __restrict__ assign, const float* __restrict__ bidv,
    const int* __restrict__ bestj,  const float* __restrict__ max_inc,
    int* __restrict__ winner)
{
  int idx = blockIdx.x * blockDim.x + threadIdx.x;
  if (idx >= BN) return;
  if (assign[idx] >= 0) return;
  float bd = bidv[idx];
  if (!(bd > NEGF)) return;
  int b = idx / NPTS, i = idx % NPTS;
  int j = bestj[idx];
  if (bd >= max_inc[b * NPTS + j]) atomicMin(&winner[b * NPTS + j], i);
}

// -------------------------------------------------------- resolve phase ----
__global__ __launch_bounds__(256) void resolve_kernel(
    int* __restrict__ winner, float* __restrict__ max_inc,
    int* __restrict__ assign, int* __restrict__ ainv, float* __restrict__ price)
{
  int idx = blockIdx.x * blockDim.x + threadIdx.x;
  if (idx >= BN) return;
  int b = idx / NPTS, j = idx % NPTS;
  int w = winner[idx];
  if (w < NPTS) {
    int old = ainv[idx];
    // evicted rows (assigned) and winner rows (unassigned) are disjoint sets,
    // and each row wins at most one column -> no write races across threads.
    if (old >= 0) assign[b * NPTS + old] = -1;
    assign[b * NPTS + w] = j;
    ainv[idx] = w;
    price[idx] += max_inc[idx];
  }
  winner[idx]  = NPTS;   // reset for next iteration
  max_inc[idx] = NEGF;
}

// ------------------------------------------------------------- finalize ----
__global__ __launch_bounds__(256) void final_kernel(
    const float* __restrict__ xyz1, const float* __restrict__ xyz2,
    const int* __restrict__ assign,
    float* __restrict__ dist_out, int* __restrict__ ass_out)
{
  int idx = blockIdx.x * blockDim.x + threadIdx.x;
  if (idx >= BN) return;
  int b = idx / NPTS;
  int a = assign[idx];
  float d = 0.0f;
  if (a >= 0) {
    float dx = xyz1[idx * 3 + 0] - xyz2[(b * NPTS + a) * 3 + 0];
    float dy = xyz1[idx * 3 + 1] - xyz2[(b * NPTS + a) * 3 + 1];
    float dz = xyz1[idx * 3 + 2] - xyz2[(b * NPTS + a) * 3 + 2];
    d = dx * dx + dy * dy + dz * dz;   // exact reference cost formula
  }
  dist_out[idx] = d;
  ass_out[idx]  = a;
}

// --------------------------------------------------------------- launch ----
extern "C" void kernel_launch(void* const* d_in, const int* in_sizes, int n_in,
                              void* d_out, int out_size, void* d_ws, size_t ws_size,
                              hipStream_t stream)
{
  const float* xyz1     = (const float*)d_in[0];
  const float* xyz2     = (const float*)d_in[1];
  // d_in[2]=ass, d_in[3]=ass_inv are ignored (reference re-inits to -1)
  const float* price_in = (const float*)d_in[4];
  const float* eps_p    = (const float*)d_in[5];
  // d_in[6]=iters is static in the reference (int(iters) == 20)

  char* ws = (char*)d_ws;
  size_t o = 0;
  float4* ypack   = (float4*)(ws + o); o += (size_t)BN * sizeof(float4); // 16B aligned first
  float*  price   = (float*) (ws + o); o += (size_t)BN * sizeof(float);
  float*  max_inc = (float*) (ws + o); o += (size_t)BN * sizeof(float);
  float*  bidv    = (float*) (ws + o); o += (size_t)BN * sizeof(float);
  int*    assign  = (int*)   (ws + o); o += (size_t)BN * sizeof(int);
  int*    ainv    = (int*)   (ws + o); o += (size_t)BN * sizeof(int);
  int*    winner  = (int*)   (ws + o); o += (size_t)BN * sizeof(int);
  int*    bestj   = (int*)   (ws + o); o += (size_t)BN * sizeof(int);
  (void)ws_size; (void)in_sizes; (void)n_in; (void)out_size;

  const int tpb     = 256;
  const int gridBN  = (BN + tpb - 1) / tpb;                 // 64 blocks
  const int gridBid = (BATCH * (NPTS / 16)) / (tpb / 32);   // 1024 waves / 8 = 128 blocks

  init_kernel<<<gridBN, tpb, 0, stream>>>(xyz2, price_in, ypack, price,
                                          max_inc, assign, ainv, winner);

  for (int it = 0; it < ITERS; ++it) {
    bid_kernel<<<gridBid, tpb, 0, stream>>>(xyz1, ypack, price, assign,
                                            bidv, bestj, max_inc, eps_p);
    winner_kernel<<<gridBN, tpb, 0, stream>>>(assign, bidv, bestj, max_inc, winner);
    resolve_kernel<<<gridBN, tpb, 0, stream>>>(winner, max_inc, assign, ainv, price);
  }

  final_kernel<<<gridBN, tpb, 0, stream>>>(xyz1, xyz2, assign,
                                           (float*)d_out, (int*)d_out + BN);
}